// StructureModule_36275293782055
// MI455X (gfx1250) — compile-verified
//
#include <hip/hip_runtime.h>
#include <math.h>

// ---------------------------------------------------------------------------
// IPA structure module for MI455X (gfx1250), wave32, V_WMMA_F32_16X16X4_F32.
// Register-blocked WMMA tiles: shared fragments across adjacent tiles to
// raise WMMA-per-VMEM density; global_prefetch for the L2-resident z pass.
// ---------------------------------------------------------------------------

typedef float v2f __attribute__((ext_vector_type(2)));
typedef float v8f __attribute__((ext_vector_type(8)));

#define Nn   512
#define NN   (512 * 512)
#define Hh   12
#define Cc   16
#define PQn  4
#define PVn  8
#define CSn  384
#define CZn  128
#define CATn 2112
#define W_C  0.14433756729740643f   /* sqrt(1/48)  */
#define W_L  0.5773502691896258f    /* sqrt(1/3)   */
#define W_P  0.13608276348795434f   /* sqrt(1/54)  */
#define INFC 100000.0f
#define EPSC 1e-8f

__device__ __forceinline__ v8f wmma4(v2f a, v2f b, v8f c) {
  // D = A(16x4,f32) * B(4x16,f32) + C(16x16,f32)
  return __builtin_amdgcn_wmma_f32_16x16x4_f32(false, a, false, b, (short)0, c,
                                               false, false);
}

__device__ __forceinline__ float softplusf(float x) {
  return logf(1.0f + expf(x));
}

// ------------------- generic WMMA GEMM, 2 M-tiles per wave ------------------
// C[M,N] = A[M,K] * B[K,N] + bias[N].  M must be a multiple of 32.
__global__ void gemm16_wmma(const float* __restrict__ A,
                            const float* __restrict__ B,
                            const float* __restrict__ bias,
                            float* __restrict__ C, int M, int N, int K) {
  int wave = blockIdx.x * (blockDim.x >> 5) + (threadIdx.x >> 5);
  int ntiles = N >> 4;
  int tiles = (M >> 5) * ntiles;
  if (wave >= tiles) return;
  int tm = (wave / ntiles) * 2, tn = wave % ntiles;
  int lane = threadIdx.x & 31;
  int m = lane & 15, hi = lane >> 4, kb = hi * 2;
  const float* A0 = A + (size_t)(tm * 16 + m) * K;
  const float* A1 = A0 + (size_t)16 * K;
  v8f acc0 = {}, acc1 = {};
  for (int k0 = 0; k0 < K; k0 += 4) {
    v2f a0, a1, b;
    a0.x = A0[k0 + kb];
    a0.y = A0[k0 + kb + 1];
    a1.x = A1[k0 + kb];
    a1.y = A1[k0 + kb + 1];
    b.x = B[(size_t)(k0 + kb) * N + tn * 16 + m];
    b.y = B[(size_t)(k0 + kb + 1) * N + tn * 16 + m];
    acc0 = wmma4(a0, b, acc0);
    acc1 = wmma4(a1, b, acc1);
  }
  float bv = bias ? bias[tn * 16 + m] : 0.0f;
#pragma unroll
  for (int r = 0; r < 8; ++r) {
    int row = tm * 16 + r + 8 * hi;
    C[(size_t)row * N + tn * 16 + m] = acc0[r] + bv;
    C[(size_t)(row + 16) * N + tn * 16 + m] = acc1[r] + bv;
  }
}

// -------- build augmented Q' (H,N,32), q-point norms (H,N) ------------------
__global__ void build_q_kernel(const float* __restrict__ q_raw,
                               const float* __restrict__ qp_raw,
                               const float* __restrict__ rot,
                               const float* __restrict__ trans,
                               const float* __restrict__ head_w,
                               float* __restrict__ Qaug,
                               float* __restrict__ qn) {
  int idx = blockIdx.x * blockDim.x + threadIdx.x;
  if (idx >= Nn * Hh) return;
  int n = idx / Hh, h = idx % Hh;
  float hw = softplusf(head_w[h]) * W_P;
  float* Qr = Qaug + ((size_t)h * Nn + n) * 32;
#pragma unroll
  for (int c = 0; c < Cc; ++c)
    Qr[c] = W_C * q_raw[n * (Hh * Cc) + h * Cc + c];
  const float* R = rot + n * 9;
  const float* t = trans + n * 3;
  float s2 = 0.0f;
#pragma unroll
  for (int d = 0; d < PQn; ++d) {
    float l0 = qp_raw[n * 144 + 0 * 48 + h * PQn + d];
    float l1 = qp_raw[n * 144 + 1 * 48 + h * PQn + d];
    float l2 = qp_raw[n * 144 + 2 * 48 + h * PQn + d];
    float p0 = R[0] * l0 + R[1] * l1 + R[2] * l2 + t[0];
    float p1 = R[3] * l0 + R[4] * l1 + R[5] * l2 + t[1];
    float p2 = R[6] * l0 + R[7] * l1 + R[8] * l2 + t[2];
    s2 += p0 * p0 + p1 * p1 + p2 * p2;
    Qr[16 + d * 3 + 0] = hw * p0;
    Qr[16 + d * 3 + 1] = hw * p1;
    Qr[16 + d * 3 + 2] = hw * p2;
  }
  Qr[28] = Qr[29] = Qr[30] = Qr[31] = 0.0f;
  qn[h * Nn + n] = s2;
}

// -------- build K' (H,32,N), k-point norms, Vcat (H,N,48) -------------------
__global__ void build_kv_kernel(const float* __restrict__ kv_raw,
                                const float* __restrict__ kvp_raw,
                                const float* __restrict__ rot,
                                const float* __restrict__ trans,
                                float* __restrict__ Kaug,
                                float* __restrict__ kn,
                                float* __restrict__ Vcat) {
  int idx = blockIdx.x * blockDim.x + threadIdx.x;
  if (idx >= Nn * Hh) return;
  int n = idx / Hh, h = idx % Hh;
  float* Vr = Vcat + ((size_t)h * Nn + n) * 48;
#pragma unroll
  for (int c = 0; c < Cc; ++c) {
    Kaug[((size_t)h * 32 + c) * Nn + n] = kv_raw[n * 384 + h * 32 + c];      // k
    Vr[c] = kv_raw[n * 384 + h * 32 + 16 + c];                               // v
  }
  const float* R = rot + n * 9;
  const float* t = trans + n * 3;
  float s2 = 0.0f;
#pragma unroll
  for (int d = 0; d < PQn + PVn; ++d) {
    float l0 = kvp_raw[n * 432 + 0 * 144 + h * 12 + d];
    float l1 = kvp_raw[n * 432 + 1 * 144 + h * 12 + d];
    float l2 = kvp_raw[n * 432 + 2 * 144 + h * 12 + d];
    float p0 = R[0] * l0 + R[1] * l1 + R[2] * l2 + t[0];
    float p1 = R[3] * l0 + R[4] * l1 + R[5] * l2 + t[1];
    float p2 = R[6] * l0 + R[7] * l1 + R[8] * l2 + t[2];
    if (d < PQn) {
      s2 += p0 * p0 + p1 * p1 + p2 * p2;
      Kaug[((size_t)h * 32 + 16 + d * 3 + 0) * Nn + n] = p0;
      Kaug[((size_t)h * 32 + 16 + d * 3 + 1) * Nn + n] = p1;
      Kaug[((size_t)h * 32 + 16 + d * 3 + 2) * Nn + n] = p2;
    } else {
      int p = d - PQn;
      Vr[16 + p * 3 + 0] = p0;
      Vr[16 + p * 3 + 1] = p1;
      Vr[16 + p * 3 + 2] = p2;
    }
  }
#pragma unroll
  for (int c = 28; c < 32; ++c) Kaug[((size_t)h * 32 + c) * Nn + n] = 0.0f;
#pragma unroll
  for (int c = 40; c < 48; ++c) Vr[c] = 0.0f;
  kn[h * Nn + n] = s2;
}

// -------- z pass 1: att[h,i,j] = w_l*(z[i,j,:]@w_b[:,h] + b_b[h]) -----------
// 4 M-tiles per wave sharing the (tiny) w_b fragment.
__global__ void bias_z_kernel(const float* __restrict__ z,
                              const float* __restrict__ w_b,
                              const float* __restrict__ b_b,
                              float* __restrict__ att) {
  int wave = blockIdx.x * (blockDim.x >> 5) + (threadIdx.x >> 5);
  if (wave >= (NN >> 6)) return;
  int row0 = wave * 64;
  int lane = threadIdx.x & 31;
  int m = lane & 15, hi = lane >> 4, kb = hi * 2;
  const float* Zr0 = z + (size_t)(row0 + m) * CZn;
  v8f acc[4] = {};
  for (int k0 = 0; k0 < CZn; k0 += 4) {
    v2f b;
    b.x = (m < Hh) ? w_b[(k0 + kb) * Hh + m] : 0.0f;
    b.y = (m < Hh) ? w_b[(k0 + kb + 1) * Hh + m] : 0.0f;
#pragma unroll
    for (int u = 0; u < 4; ++u) {
      v2f a;
      const float* Zr = Zr0 + (size_t)u * 16 * CZn;
      a.x = Zr[k0 + kb];
      a.y = Zr[k0 + kb + 1];
      acc[u] = wmma4(a, b, acc[u]);
    }
  }
  if (m < Hh) {
    float bb = b_b[m];
#pragma unroll
    for (int u = 0; u < 4; ++u)
#pragma unroll
      for (int r = 0; r < 8; ++r) {
        int row = row0 + u * 16 + r + 8 * hi;
        att[(size_t)m * NN + row] = W_L * (acc[u][r] + bb);
      }
  }
}

// -------- logits: att += Q'[h] @ K'[h] - 0.5*hw*(qn_i + kn_j) ---------------
// 4 j-tiles per wave sharing the Q' fragment.
__global__ void logits_kernel(const float* __restrict__ Qaug,
                              const float* __restrict__ Kaug,
                              const float* __restrict__ qn,
                              const float* __restrict__ kn,
                              const float* __restrict__ head_w,
                              float* __restrict__ att) {
  int wave = blockIdx.x * (blockDim.x >> 5) + (threadIdx.x >> 5);
  if (wave >= Hh * 32 * 8) return;
  int h = wave / 256;
  int rem = wave % 256;
  int ti = rem / 8, tj0 = (rem % 8) * 4;
  int lane = threadIdx.x & 31;
  int m = lane & 15, hi = lane >> 4, kb = hi * 2;
  const float* Ah = Qaug + (size_t)h * Nn * 32 + (size_t)(ti * 16 + m) * 32;
  const float* Bh = Kaug + (size_t)h * 32 * Nn;
  v8f acc[4] = {};
#pragma unroll
  for (int k0 = 0; k0 < 32; k0 += 4) {
    v2f a;
    a.x = Ah[k0 + kb];
    a.y = Ah[k0 + kb + 1];
    const float* Br0 = Bh + (size_t)(k0 + kb) * Nn + tj0 * 16 + m;
    const float* Br1 = Bh + (size_t)(k0 + kb + 1) * Nn + tj0 * 16 + m;
#pragma unroll
    for (int u = 0; u < 4; ++u) {
      v2f b;
      b.x = Br0[u * 16];
      b.y = Br1[u * 16];
      acc[u] = wmma4(a, b, acc[u]);
    }
  }
  float hw = softplusf(head_w[h]) * W_P;
#pragma unroll
  for (int u = 0; u < 4; ++u) {
    int j = (tj0 + u) * 16 + m;
    float knj = kn[h * Nn + j];
#pragma unroll
    for (int r = 0; r < 8; ++r) {
      int i = ti * 16 + r + 8 * hi;
      size_t idx = (size_t)h * NN + (size_t)i * Nn + j;
      att[idx] = att[idx] + acc[u][r] - 0.5f * hw * (qn[h * Nn + i] + knj);
    }
  }
}

// -------- softmax over j for each (h,i) row --------------------------------
__global__ void softmax_kernel(float* __restrict__ att,
                               const float* __restrict__ mask) {
  int row = blockIdx.x;             // h*512 + i
  int i = row % Nn;
  float* p = att + (size_t)row * Nn;
  float mi = mask[i];
  __shared__ float red[256];
  int t = threadIdx.x;
  float v0 = p[t] + INFC * (mi * mask[t] - 1.0f);
  float v1 = p[t + 256] + INFC * (mi * mask[t + 256] - 1.0f);
  red[t] = fmaxf(v0, v1);
  __syncthreads();
  for (int s = 128; s > 0; s >>= 1) {
    if (t < s) red[t] = fmaxf(red[t], red[t + s]);
    __syncthreads();
  }
  float mx = red[0];
  __syncthreads();
  float e0 = expf(v0 - mx), e1 = expf(v1 - mx);
  red[t] = e0 + e1;
  __syncthreads();
  for (int s = 128; s > 0; s >>= 1) {
    if (t < s) red[t] += red[t + s];
    __syncthreads();
  }
  float inv = 1.0f / red[0];
  p[t] = e0 * inv;
  p[t + 256] = e1 * inv;
}

// -------- o / o_pt: att[h] (512x512) @ Vcat[h] (512x48) ---------------------
// All 3 N-tiles per wave sharing the attention fragment.
__global__ void out_gemm_kernel(const float* __restrict__ att,
                                const float* __restrict__ Vcat,
                                float* __restrict__ catb,
                                float* __restrict__ opre) {
  int wave = blockIdx.x * (blockDim.x >> 5) + (threadIdx.x >> 5);
  if (wave >= Hh * 32) return;
  int h = wave / 32;
  int tm = wave % 32;
  int lane = threadIdx.x & 31;
  int m = lane & 15, hi = lane >> 4, kb = hi * 2;
  const float* A = att + (size_t)h * NN + (size_t)(tm * 16 + m) * Nn;
  const float* B = Vcat + (size_t)h * Nn * 48;
  v8f acc[3] = {};
  for (int k0 = 0; k0 < Nn; k0 += 4) {
    v2f a;
    a.x = A[k0 + kb];
    a.y = A[k0 + kb + 1];
    const float* Br0 = B + (size_t)(k0 + kb) * 48 + m;
    const float* Br1 = B + (size_t)(k0 + kb + 1) * 48 + m;
#pragma unroll
    for (int u = 0; u < 3; ++u) {
      v2f b;
      b.x = Br0[u * 16];
      b.y = Br1[u * 16];
      acc[u] = wmma4(a, b, acc[u]);
    }
  }
#pragma unroll
  for (int u = 0; u < 3; ++u) {
    int ch = u * 16 + m;
#pragma unroll
    for (int r = 0; r < 8; ++r) {
      int i = tm * 16 + r + 8 * hi;
      if (ch < 16) {
        catb[(size_t)i * CATn + h * Cc + ch] = acc[u][r];      // o section
      } else if (ch < 40) {
        int pc = ch - 16;
        opre[(((size_t)i * Hh + h) * PVn + pc / 3) * 3 + pc % 3] = acc[u][r];
      }
    }
  }
}

// -------- invariant-frame finalize of o_pt + norms --------------------------
__global__ void finalize_pts_kernel(const float* __restrict__ opre,
                                    const float* __restrict__ rot,
                                    const float* __restrict__ trans,
                                    float* __restrict__ catb) {
  int idx = blockIdx.x * blockDim.x + threadIdx.x;
  if (idx >= Nn * Hh) return;
  int n = idx / Hh, h = idx % Hh;
  const float* R = rot + n * 9;
  const float* t = trans + n * 3;
  float* base = catb + (size_t)n * CATn + Hh * Cc;   // start of o_pt.x section
#pragma unroll
  for (int p = 0; p < PVn; ++p) {
    const float* op = opre + (((size_t)n * Hh + h) * PVn + p) * 3;
    float x0 = op[0] - t[0], x1 = op[1] - t[1], x2 = op[2] - t[2];
    float y0 = R[0] * x0 + R[3] * x1 + R[6] * x2;    // R^T * x
    float y1 = R[1] * x0 + R[4] * x1 + R[7] * x2;
    float y2 = R[2] * x0 + R[5] * x1 + R[8] * x2;
    float nrm = sqrtf(y0 * y0 + y1 * y1 + y2 * y2 + EPSC);
    int c = h * PVn + p;
    base[c] = y0;
    base[96 + c] = y1;
    base[192 + c] = y2;
    base[288 + c] = nrm;
  }
}

// -------- z pass 2 (L2-resident): o_pair[i] = att[:,i,:] @ z[i] -------------
// 4 N-tiles per wave sharing the attention fragment; prefetch the z stream.
__global__ void pair_gemm_kernel(const float* __restrict__ att,
                                 const float* __restrict__ z,
                                 float* __restrict__ catb) {
  int wave = blockIdx.x * (blockDim.x >> 5) + (threadIdx.x >> 5);
  if (wave >= Nn * 2) return;
  int i = wave >> 1, ng = (wave & 1) * 64;   // 64-column group
  int lane = threadIdx.x & 31;
  int m = lane & 15, hi = lane >> 4, kb = hi * 2;
  v8f acc[4] = {};
  for (int k0 = 0; k0 < Nn; k0 += 4) {
    v2f a;
    a.x = (m < Hh) ? att[(size_t)m * NN + (size_t)i * Nn + k0 + kb] : 0.0f;
    a.y = (m < Hh) ? att[(size_t)m * NN + (size_t)i * Nn + k0 + kb + 1] : 0.0f;
    const float* Zr0 = z + ((size_t)i * Nn + k0 + kb) * CZn + ng;
    const float* Zr1 = Zr0 + CZn;
    // Pull the upcoming z rows toward the WGP (z is L2-resident after pass 1).
    __builtin_prefetch(Zr0 + 16 * CZn, 0, 1);
#pragma unroll
    for (int u = 0; u < 4; ++u) {
      v2f b;
      b.x = Zr0[u * 16 + m];
      b.y = Zr1[u * 16 + m];
      acc[u] = wmma4(a, b, acc[u]);
    }
  }
#pragma unroll
  for (int u = 0; u < 4; ++u)
#pragma unroll
    for (int r = 0; r < 8; ++r) {
      int h = r + 8 * hi;
      if (h < Hh)
        catb[(size_t)i * CATn + 576 + h * CZn + ng + u * 16 + m] = acc[u][r];
    }
}

// ---------------------------------------------------------------------------
extern "C" void kernel_launch(void* const* d_in, const int* in_sizes, int n_in,
                              void* d_out, int out_size, void* d_ws,
                              size_t ws_size, hipStream_t stream) {
  (void)in_sizes; (void)n_in; (void)out_size; (void)ws_size;
  const float* s      = (const float*)d_in[0];
  const float* z      = (const float*)d_in[1];
  const float* rot    = (const float*)d_in[2];
  const float* trans  = (const float*)d_in[3];
  const float* mask   = (const float*)d_in[4];
  const float* w_q    = (const float*)d_in[5];
  const float* b_q    = (const float*)d_in[6];
  const float* w_kv   = (const float*)d_in[7];
  const float* b_kv   = (const float*)d_in[8];
  const float* w_qp   = (const float*)d_in[9];
  const float* b_qp   = (const float*)d_in[10];
  const float* w_kvp  = (const float*)d_in[11];
  const float* b_kvp  = (const float*)d_in[12];
  const float* w_b    = (const float*)d_in[13];
  const float* b_b    = (const float*)d_in[14];
  const float* head_w = (const float*)d_in[15];
  const float* w_out  = (const float*)d_in[16];
  const float* b_out  = (const float*)d_in[17];
  float* out = (float*)d_out;

  float* ws = (float*)d_ws;
  float* q_raw   = ws; ws += Nn * 192;
  float* kv_raw  = ws; ws += Nn * 384;
  float* qp_raw  = ws; ws += Nn * 144;
  float* kvp_raw = ws; ws += Nn * 432;
  float* Qaug    = ws; ws += Hh * Nn * 32;
  float* Kaug    = ws; ws += Hh * Nn * 32;
  float* Vcat    = ws; ws += Hh * Nn * 48;
  float* qn      = ws; ws += Hh * Nn;
  float* kn      = ws; ws += Hh * Nn;
  float* att     = ws; ws += (size_t)Hh * NN;
  float* opre    = ws; ws += Nn * Hh * PVn * 3;
  float* catb    = ws; ws += Nn * CATn;

  auto gemm = [&](const float* A, const float* B, const float* bias, float* C,
                  int M, int N, int K) {
    int tiles = (M / 32) * (N / 16);
    gemm16_wmma<<<(tiles + 7) / 8, 256, 0, stream>>>(A, B, bias, C, M, N, K);
  };

  // 1) projections
  gemm(s, w_q,   b_q,   q_raw,   Nn, 192, CSn);
  gemm(s, w_kv,  b_kv,  kv_raw,  Nn, 384, CSn);
  gemm(s, w_qp,  b_qp,  qp_raw,  Nn, 144, CSn);
  gemm(s, w_kvp, b_kvp, kvp_raw, Nn, 432, CSn);

  // 2) build augmented operands + frames
  build_q_kernel<<<(Nn * Hh + 255) / 256, 256, 0, stream>>>(
      q_raw, qp_raw, rot, trans, head_w, Qaug, qn);
  build_kv_kernel<<<(Nn * Hh + 255) / 256, 256, 0, stream>>>(
      kv_raw, kvp_raw, rot, trans, Kaug, kn, Vcat);

  // 3) z pass 1: pair bias -> logits buffer (transposed, pre-scaled by w_l)
  bias_z_kernel<<<((NN / 64) + 7) / 8, 256, 0, stream>>>(z, w_b, b_b, att);

  // 4) attention logits (q·k + point attention via augmented GEMM)
  logits_kernel<<<(Hh * 32 * 8 + 7) / 8, 256, 0, stream>>>(
      Qaug, Kaug, qn, kn, head_w, att);

  // 5) softmax over j
  softmax_kernel<<<Hh * Nn, 256, 0, stream>>>(att, mask);

  // 6) o and raw o_pt
  out_gemm_kernel<<<(Hh * 32 + 7) / 8, 256, 0, stream>>>(
      att, Vcat, catb, opre);

  // 7) invariant-frame transform + norms
  finalize_pts_kernel<<<(Nn * Hh + 255) / 256, 256, 0, stream>>>(
      opre, rot, trans, catb);

  // 8) z pass 2: o_pair (z should largely hit the 192MB L2 here)
  pair_gemm_kernel<<<(Nn * 2 + 7) / 8, 256, 0, stream>>>(att, z, catb);

  // 9) final projection
  gemm(catb, w_out, b_out, out, Nn, CSn, CATn);
}